// TopLoss_4415226380588
// MI455X (gfx1250) — compile-verified
//
#include <hip/hip_runtime.h>
#include <stdint.h>

// ---------------- problem constants (128x128 grid, Freudenthal) ----------------
#define N0 128
#define N1 128
#define NV (N0*N1)          // 16384 vertices
#define EH 16256            // horizontal edges 128*127
#define EV 16256            // vertical edges 127*128
#define ED 16129            // diagonal edges 127*127
#define NE (EH+EV+ED)       // 48641 primal edges == dual edges
#define NSORT 65536         // pow2 pad for bitonic sort
#define NLOW 16129          // lower triangles
#define NT 32258            // total triangles
#define OUTER_FACE NT       // 32258
#define P1BASE 48387        // 3*NLOW : first pass-2 dual slot

// ---------------- order-preserving float <-> uint32 ----------------
__device__ __forceinline__ uint32_t f2ord(float x) {
  uint32_t b = __float_as_uint(x);
  return b ^ (uint32_t)(((int32_t)b >> 31) | (int32_t)0x80000000);
}
__device__ __forceinline__ float ord2f(uint32_t o) {
  uint32_t b = (o & 0x80000000u) ? (o ^ 0x80000000u) : ~o;
  return __uint_as_float(b);
}

// ---------------- primal edge id -> (u,v), u < v ----------------
__device__ __forceinline__ void edge0_uv(uint32_t e, uint32_t& u, uint32_t& v) {
  if (e < EH)            { uint32_t i = e / 127u, j = e - 127u*i; u = (i<<7)+j; v = u+1u;   }
  else if (e < EH+EV)    { uint32_t h = e - EH; uint32_t i = h >> 7, j = h & 127u; u = (i<<7)+j; v = u+128u; }
  else                   { uint32_t d = e - (EH+EV); uint32_t i = d / 127u, j = d - 127u*i; u = (i<<7)+j; v = u+129u; }
}

// ---------------- dual-edge slot -> primal (u,v) of its shared edge ----------------
// Slot order replicates Python dict insertion order of edge2face:
//   s = 3t+r for lower triangle t (r=0 vertical(i,j), r=1 diagonal(i,j), r=2 horizontal(i+1,j))
//   then p = s-P1BASE: p<=126 -> horizontal(0,p) ; p>=127 -> vertical(p-127, 127)
__device__ __forceinline__ void dual_uv(uint32_t s, uint32_t& u, uint32_t& v) {
  if (s < P1BASE) {
    uint32_t t = s / 3u, r = s - 3u*t;
    uint32_t i = t / 127u, j = t - 127u*i;
    uint32_t base = (i<<7) + j;
    if (r == 0u)      { u = base;        v = base + 128u; }   // vertical (i..i+1, j)
    else if (r == 1u) { u = base;        v = base + 129u; }   // diagonal cell (i,j)
    else              { u = base + 128u; v = base + 129u; }   // horizontal (i+1, j..j+1)
  } else {
    uint32_t p = s - P1BASE;
    if (p <= 126u) { u = p; v = p + 1u; }                      // horizontal row 0
    else { uint32_t i = p - 127u; u = (i<<7) + 127u; v = u + 128u; } // vertical col 127
  }
}

// ---------------- dual-edge slot -> (du, dv) face ids ----------------
__device__ __forceinline__ void dual_faces(uint32_t s, uint32_t& du, uint32_t& dv) {
  if (s < P1BASE) {
    uint32_t t = s / 3u, r = s - 3u*t;
    uint32_t i = t / 127u, j = t - 127u*i;
    du = t;                                            // lower triangle inserts first
    if (r == 0u)      dv = j ? (NLOW + i*127u + (j-1u)) : OUTER_FACE;
    else if (r == 1u) dv = NLOW + t;
    else              dv = (i < 126u) ? (NLOW + (i+1u)*127u + j) : OUTER_FACE;
  } else {
    uint32_t p = s - P1BASE;
    du = (p <= 126u) ? (NLOW + p) : (NLOW + (p - 127u)*127u + 126u);
    dv = OUTER_FACE;                                   // all pass-2 edges are boundary
  }
}

// ---------------- Tensor Data Mover: stage 2D f32 tile global -> LDS ----------------
typedef uint32_t v4u __attribute__((ext_vector_type(4)));
typedef int      v8i __attribute__((ext_vector_type(8)));
typedef int      v4i __attribute__((ext_vector_type(4)));

__device__ __forceinline__ void tdm_load_2d_f32(const void* gptr, uint32_t lds_off,
                                                uint32_t width, uint32_t height) {
#if defined(__HIP_DEVICE_COMPILE__)
  uint64_t ga = (uint64_t)(uintptr_t)gptr;
  v4u g0;
  g0.x = 1u;                                                // count=1, user descriptor
  g0.y = lds_off;                                           // lds_addr (bytes)
  g0.z = (uint32_t)ga;                                      // global_addr[31:0]
  g0.w = (uint32_t)((ga >> 32) & 0x01FFFFFFull) | (2u << 30); // addr[56:32] | type=2
  v8i g1;
  g1[0] = (int)(2u << 16);                                  // data_size = 4B
  g1[1] = (int)((width & 0xFFFFu) << 16);                   // tensor_dim0[15:0] @ bits63:48
  g1[2] = (int)(((width >> 16) & 0xFFFFu) | ((height & 0xFFFFu) << 16)); // dim0_hi | dim1_lo
  g1[3] = (int)(((height >> 16) & 0xFFFFu) | ((width & 0xFFFFu) << 16)); // dim1_hi | tile_dim0
  g1[4] = (int)(height & 0xFFFFu);                          // tile_dim1 | tile_dim2=0
  g1[5] = (int)width;                                       // tensor_dim0_stride lo32
  g1[6] = 0;                                                // stride hi | dim1_stride lo
  g1[7] = 0;
  v4i z4 = {0, 0, 0, 0};
  v8i z8 = {0, 0, 0, 0, 0, 0, 0, 0};
  // 6-arg clang-23 form: (g0, g1, group2, group3, group4, cpol)
  __builtin_amdgcn_tensor_load_to_lds(g0, g1, z4, z4, z8, 0);
#else
  (void)gptr; (void)lds_off; (void)width; (void)height;
#endif
}
__device__ __forceinline__ void tdm_wait_all() {
#if defined(__HIP_DEVICE_COMPILE__)
  __builtin_amdgcn_s_wait_tensorcnt(0);
#endif
}

// ---------------- kernel 1: build sort keys for both filtrations ----------------
// A[t] ascending-sort key of primal edge t: (ord(max(f_u,f_v)) << 16) | t   (f = -beta)
// B[s] ascending-sort key of dual edge s : (~ord(dval) << 16) | s          (== descending by dval)
__global__ void build_keys(const float* __restrict__ beta,
                           uint64_t* __restrict__ A, uint64_t* __restrict__ B) {
  uint32_t t = blockIdx.x * blockDim.x + threadIdx.x;
  if (t < NSORT) {
    uint64_t key = ~0ull;
    if (t < NE) {
      uint32_t u, v; edge0_uv(t, u, v);
      float val = fmaxf(-beta[u], -beta[v]);
      key = ((uint64_t)f2ord(val) << 16) | (uint64_t)t;
    }
    A[t] = key;
  } else {
    uint32_t s = t - NSORT;
    uint64_t key = ~0ull;
    if (s < NE) {
      uint32_t u, v; dual_uv(s, u, v);
      float val = fmaxf(-beta[u], -beta[v]);
      key = ((uint64_t)(uint32_t)(~f2ord(val)) << 16) | (uint64_t)s;
    }
    B[s] = key;
  }
}

// ---------------- kernel 2: one bitonic compare-exchange pass over BOTH arrays ----------------
__global__ void bitonic_pass(uint64_t* __restrict__ A, uint64_t* __restrict__ B,
                             uint32_t j, uint32_t k) {
  uint32_t t = blockIdx.x * blockDim.x + threadIdx.x;     // 0 .. 2*NSORT-1
  uint64_t* arr = (t & NSORT) ? B : A;
  uint32_t i = t & (NSORT - 1u);
  uint32_t p = i ^ j;
  if (p > i) {
    uint64_t a = arr[i], b = arr[p];
    bool asc = ((i & k) == 0u);
    if ((a > b) == asc) { arr[i] = b; arr[p] = a; }
  }
}

// ---------------- kernel 3: dim-0 Kruskal union-find in LDS ----------------
// LDS: [0,64K) beta stage | [64K,96K) parent u16 | [96K,160K) birth f32
__global__ void uf_dim0(const float* __restrict__ beta,
                        const uint64_t* __restrict__ A,
                        double* __restrict__ loss0_out) {
  extern __shared__ __align__(16) uint8_t smem[];
  float*    stage  = (float*)smem;
  uint16_t* parent = (uint16_t*)(smem + 65536);
  float*    birth  = (float*)(smem + 98304);

  if (threadIdx.x == 0)
    tdm_load_2d_f32(beta, (uint32_t)(uintptr_t)(void*)stage, N1, N0);
  tdm_wait_all();
  __syncthreads();

  for (uint32_t i = threadIdx.x; i < NV; i += blockDim.x) {
    parent[i] = (uint16_t)i;
    birth[i]  = -stage[i];           // f = -beta ; singleton birth value
  }
  __syncthreads();

  if (threadIdx.x == 0) {
    double loss0 = 0.0;              // = sum f[death] - f[birth] over dim-0 pairs
    for (uint32_t e = 0; e < NE; ++e) {
      uint64_t k   = A[e];
      uint32_t idx = (uint32_t)(k & 0xFFFFu);
      float    val = ord2f((uint32_t)(k >> 16));
      uint32_t u, v; edge0_uv(idx, u, v);
      uint32_t ru = u;               // find with path halving
      for (;;) { uint32_t p = parent[ru]; if (p == ru) break;
                 uint32_t g = parent[p]; parent[ru] = (uint16_t)g; ru = g; }
      uint32_t rv = v;
      for (;;) { uint32_t p = parent[rv]; if (p == rv) break;
                 uint32_t g = parent[p]; parent[rv] = (uint16_t)g; rv = g; }
      if (ru == rv) continue;
      float bu = birth[ru], bv = birth[rv];
      if (bu < bv) { uint32_t tr = ru; ru = rv; rv = tr; float tb = bu; bu = bv; bv = tb; }
      loss0 += (double)val - (double)bu;   // younger (larger birth f) dies
      parent[ru] = (uint16_t)rv;           // survivor keeps older (smaller) birth
    }
    loss0_out[0] = loss0;
  }
}

// ---------------- kernel 4: dim-1 union-find on dual graph, final output ----------------
// LDS: [0,64K) beta stage | [64K,128K) parent u16 (32259) | [128K,...) birth f32 (32259)
__global__ void uf_dim1(const float* __restrict__ beta,
                        const uint64_t* __restrict__ B,
                        const double* __restrict__ loss0_in,
                        float* __restrict__ out) {
  extern __shared__ __align__(16) uint8_t smem[];
  float*    stage  = (float*)smem;
  uint16_t* parent = (uint16_t*)(smem + 65536);
  float*    birth  = (float*)(smem + 131072);

  if (threadIdx.x == 0)
    tdm_load_2d_f32(beta, (uint32_t)(uintptr_t)(void*)stage, N1, N0);
  tdm_wait_all();
  __syncthreads();

  for (uint32_t t = threadIdx.x; t < NT; t += blockDim.x) {
    uint32_t a, b, c;
    if (t < NLOW) { uint32_t i = t/127u,        j = t - 127u*i;        a=(i<<7)+j; b=a+128u; c=a+129u; }
    else          { uint32_t tt = t-NLOW; uint32_t i = tt/127u, j = tt-127u*i;  a=(i<<7)+j; b=a+1u;   c=a+129u; }
    float m  = -stage[a];
    float fb = -stage[b]; if (fb > m) m = fb;
    float fc = -stage[c]; if (fc > m) m = fc;   // triangle birth value f[t_crit]
    parent[t] = (uint16_t)t;
    birth[t]  = m;
  }
  if (threadIdx.x == 0) { parent[OUTER_FACE] = (uint16_t)OUTER_FACE; birth[OUTER_FACE] = __builtin_inff(); }
  __syncthreads();

  if (threadIdx.x == 0) {
    double sum1 = 0.0;
    float  max1 = -__builtin_inff();
    for (uint32_t e = 0; e < NE; ++e) {
      uint64_t k = B[e];
      uint32_t s = (uint32_t)(k & 0xFFFFu);
      float  val = ord2f(~(uint32_t)(k >> 16));      // dval = f[dcrit]
      uint32_t du, dv; dual_faces(s, du, dv);
      uint32_t ru = du;
      for (;;) { uint32_t p = parent[ru]; if (p == ru) break;
                 uint32_t g = parent[p]; parent[ru] = (uint16_t)g; ru = g; }
      uint32_t rv = dv;
      for (;;) { uint32_t p = parent[rv]; if (p == rv) break;
                 uint32_t g = parent[p]; parent[rv] = (uint16_t)g; rv = g; }
      if (ru == rv) continue;
      float bu = birth[ru], bv = birth[rv];
      if (bu > bv) { uint32_t tr = ru; ru = rv; rv = tr; float tb = bu; bu = bv; bv = tb; }
      float l = bu - val;                 // bar length beta[be1]-beta[dt1]
      sum1 += (double)l;
      if (l > max1) max1 = l;
      parent[ru] = (uint16_t)rv;          // younger (smaller max face) dies
    }
    out[0] = (float)(loss0_in[0] + sum1 - (double)max1);  // skip the longest dim-1 bar
  }
}

// ---------------- host entry ----------------
extern "C" void kernel_launch(void* const* d_in, const int* in_sizes, int n_in,
                              void* d_out, int out_size, void* d_ws, size_t ws_size,
                              hipStream_t stream) {
  (void)in_sizes; (void)n_in; (void)out_size; (void)ws_size;
  const float* beta = (const float*)d_in[0];
  uint64_t* A = (uint64_t*)d_ws;           // 65536 * 8 B
  uint64_t* B = A + NSORT;                 // 65536 * 8 B
  double* loss0 = (double*)(B + NSORT);    // 8 B scratch
  float* out = (float*)d_out;

  // Allow >64KB dynamic LDS for the union-find kernels (CDNA5 WGP has 320KB).
  (void)hipFuncSetAttribute(reinterpret_cast<const void*>(uf_dim0),
                            hipFuncAttributeMaxDynamicSharedMemorySize, 163840);
  (void)hipFuncSetAttribute(reinterpret_cast<const void*>(uf_dim1),
                            hipFuncAttributeMaxDynamicSharedMemorySize, 260112);

  build_keys<<<(2 * NSORT) / 256, 256, 0, stream>>>(beta, A, B);

  for (uint32_t k = 2; k <= NSORT; k <<= 1)
    for (uint32_t j = k >> 1; j > 0; j >>= 1)
      bitonic_pass<<<(2 * NSORT) / 256, 256, 0, stream>>>(A, B, j, k);

  uf_dim0<<<1, 256, 163840, stream>>>(beta, A, loss0);
  uf_dim1<<<1, 256, 260112, stream>>>(beta, B, loss0, out);
}